// NonGalerkinLoss1Rand_87041807221105
// MI455X (gfx1250) — compile-verified
//
#include <hip/hip_runtime.h>
#include <hip/hip_bf16.h>

// ---------------------------------------------------------------------------
// NonGalerkinLoss1Rand for MI455X (gfx1250, wave32).
//
// Key observation: phmz has support only on columns 4080..4112 (33 cols), so
// the big (1024x8192)@(8192x33) product collapses to a (1024x33)@(33x33) GEMM.
// Everything is fp32 and latency-bound; we run the GEMM on the fp32 WMMA path
// (V_WMMA_F32_16X16X4_F32) to keep reference precision.
//
// Workspace layout (floats):
//   Acol   [36][1024]  : stencil coefficients, column-major (K padded 33->36)
//   Ypad   [36][48]    : Ypad[j][k] = ykinput[k][4080+j]   (N padded 33->48)
//   part   [192]       : per-wave loss partials (64 mtiles x 3 ntiles)
// ---------------------------------------------------------------------------

typedef __attribute__((ext_vector_type(2))) float v2f;
typedef __attribute__((ext_vector_type(8))) float v8f;

#define VEC     8192
#define CNTR    4096
#define BATCH   1024
#define KPAD    36          // 33 -> 36, 9 WMMA k-steps of 4
#define NPAD    48          // 33 -> 48, 3 N tiles of 16
#define MTILES  (BATCH/16)  // 64
#define NTILES  (NPAD/16)   // 3
#define NWAVES  (MTILES*NTILES)

// --- kernel 1: per-batch 3-tap stencil, m-1 applications on a 33-wide window
__global__ __launch_bounds__(256) void ngl_stencil(const float* __restrict__ input,
                                                   float* __restrict__ Acol) {
    const int b = blockIdx.x * blockDim.x + threadIdx.x;
    if (b >= BATCH) return;
    const float p0 = input[4 * b + 0];
    const float p1 = input[4 * b + 1];
    const float p2 = input[4 * b + 2];
    const int   m  = (int)input[4 * b + 3];   // 1..16

    // local j <-> global column 4080+j ; center 4096 -> j=16
    float cur[33];
#pragma unroll
    for (int j = 0; j < 33; ++j) cur[j] = 0.0f;
    cur[15] = p0; cur[16] = p1; cur[17] = p2;

    const int steps = m - 1;                  // inactive steps are identity
    for (int t = 0; t < steps; ++t) {
        float nxt[33];
#pragma unroll
        for (int j = 0; j < 33; ++j) {
            const float l = (j > 0)  ? cur[j - 1] : 0.0f;
            const float r = (j < 32) ? cur[j + 1] : 0.0f;
            // new[g] = p0*old[g-1] + p1*old[g] + p2*old[g+1]
            nxt[j] = fmaf(p0, l, fmaf(p1, cur[j], p2 * r));
        }
#pragma unroll
        for (int j = 0; j < 33; ++j) cur[j] = nxt[j];
    }

    // column-major store -> coalesced per j iteration, coalesced WMMA A loads
#pragma unroll
    for (int j = 0; j < 33; ++j) Acol[j * BATCH + b] = cur[j];
    Acol[33 * BATCH + b] = 0.0f;
    Acol[34 * BATCH + b] = 0.0f;
    Acol[35 * BATCH + b] = 0.0f;
}

// --- kernel 2: pack B operand Ypad[j][k] = ykinput[k][4080+j], zero padded
__global__ __launch_bounds__(256) void ngl_prep_y(const float* __restrict__ yk,
                                                  float* __restrict__ Ypad) {
    for (int idx = blockIdx.x * blockDim.x + threadIdx.x; idx < KPAD * NPAD;
         idx += blockDim.x * gridDim.x) {
        const int j = idx / NPAD;
        const int k = idx % NPAD;
        float v = 0.0f;
        if (j < 33 && k < 33) v = yk[k * VEC + (CNTR - 16 + j)];
        Ypad[idx] = v;
    }
}

// --- kernel 3: one wave per 16x16 output tile; 9x f32 WMMA + fused loss epilogue
__global__ __launch_bounds__(32) void ngl_gemm_loss(const float* __restrict__ Acol,
                                                    const float* __restrict__ Ypad,
                                                    const float* __restrict__ outp,
                                                    const float* __restrict__ yk,
                                                    float* __restrict__ part) {
    const int lane   = threadIdx.x;     // 0..31 (wave32)
    const int m_base = blockIdx.x * 16; // 0..1008
    const int n_base = blockIdx.y * 16; // 0, 16, 32
    const int half   = lane >> 4;       // ISA f32 A/B layout: lanes 16-31 hold K+2,K+3
    const int mloc   = lane & 15;
    const int nloc   = lane & 15;

    v8f acc = {};
#pragma unroll
    for (int kk = 0; kk < 9; ++kk) {
        const int k0 = kk * 4 + half * 2;
        v2f a, bmat;
        // A 16x4 f32: lane holds A[mloc][k0], A[mloc][k0+1]  (16 coalesced lanes)
        a.x    = Acol[(k0    ) * BATCH + m_base + mloc];
        a.y    = Acol[(k0 + 1) * BATCH + m_base + mloc];
        // B 4x16 f32: lane holds B[k0][nloc], B[k0+1][nloc]
        bmat.x = Ypad[(k0    ) * NPAD + n_base + nloc];
        bmat.y = Ypad[(k0 + 1) * NPAD + n_base + nloc];
        // D = A*B + C  (fp32 matrix pipe, RNE, full precision)
        acc = __builtin_amdgcn_wmma_f32_16x16x4_f32(
            /*neg_a=*/false, a, /*neg_b=*/false, bmat,
            /*c_mod=*/(short)0, acc, /*reuse_a=*/false, /*reuse_b=*/false);
    }

    // Epilogue: D VGPR v holds phim[m_base + v + 8*half][n_base + nloc]
    const int n_g = n_base + nloc;
    float lsum = 0.0f;
    if (n_g < 33) {
        const float y0 = yk[n_g * VEC + (CNTR - 1)];
        const float y1 = yk[n_g * VEC + (CNTR    )];
        const float y2 = yk[n_g * VEC + (CNTR + 1)];
#pragma unroll
        for (int v = 0; v < 8; ++v) {
            const int   m_g  = m_base + v + half * 8;
            const float o0   = outp[m_g * 3 + 0];
            const float o1   = outp[m_g * 3 + 1];
            const float o2   = outp[m_g * 3 + 2];
            const float psi  = fmaf(o0, y0, fmaf(o1, y1, o2 * y2));
            const float phim = acc[v];
            const float d    = phim - psi;
            const float g    = 1.0f - phim;
            lsum += (d * d) / (g * g + 0.01f);
        }
    }
    // wave32 reduction
#pragma unroll
    for (int off = 16; off >= 1; off >>= 1) lsum += __shfl_xor(lsum, off, 32);
    if (lane == 0) part[blockIdx.y * MTILES + blockIdx.x] = lsum;
}

// --- kernel 4: deterministic final reduction of 192 partials
__global__ __launch_bounds__(32) void ngl_reduce(const float* __restrict__ part,
                                                 float* __restrict__ out) {
    float s = 0.0f;
    for (int i = threadIdx.x; i < NWAVES; i += 32) s += part[i];
#pragma unroll
    for (int off = 16; off >= 1; off >>= 1) s += __shfl_xor(s, off, 32);
    if (threadIdx.x == 0) out[0] = s;
}

extern "C" void kernel_launch(void* const* d_in, const int* in_sizes, int n_in,
                              void* d_out, int out_size, void* d_ws, size_t ws_size,
                              hipStream_t stream) {
    const float* input  = (const float*)d_in[0];   // (1024,4)
    const float* output = (const float*)d_in[1];   // (1024,3)
    const float* yk     = (const float*)d_in[2];   // (33,8192)
    float*       out    = (float*)d_out;

    float* Acol = (float*)d_ws;                    // 36*1024 floats
    float* Ypad = Acol + KPAD * BATCH;             // 36*48 floats
    float* part = Ypad + KPAD * NPAD;              // 192 floats
    (void)in_sizes; (void)n_in; (void)out_size; (void)ws_size;

    ngl_stencil  <<<BATCH / 256, 256, 0, stream>>>(input, Acol);
    ngl_prep_y   <<<(KPAD * NPAD + 255) / 256, 256, 0, stream>>>(yk, Ypad);
    ngl_gemm_loss<<<dim3(MTILES, NTILES), 32, 0, stream>>>(Acol, Ypad, output, yk, part);
    ngl_reduce   <<<1, 32, 0, stream>>>(part, out);
}